// CubeSimulator_56599079026823
// MI455X (gfx1250) — compile-verified
//
#include <hip/hip_runtime.h>
#include <math.h>

#define PC_F 3.086e16f
#define LOG2E_F 1.44269504088896340736f

// ---- CDNA5 async global->LDS copy (ISA §10: GLOBAL_LOAD_ASYNC_TO_LDS_B128) ----
// Per-lane: copies 16 bytes from global address gaddr to LDS byte address lds_byte_off; tracked by ASYNCcnt.
__device__ __forceinline__ void async_g2l_b128(unsigned int lds_byte_off, unsigned long long gaddr)
{
    asm volatile("global_load_async_to_lds_b128 %0, %1, off"
                 :: "v"(lds_byte_off), "v"(gaddr)
                 : "memory");
}

__device__ __forceinline__ void wait_asynccnt0()
{
#if __has_builtin(__builtin_amdgcn_s_wait_asynccnt)
    __builtin_amdgcn_s_wait_asynccnt(0);
#else
    asm volatile("s_wait_asynccnt 0x0" ::: "memory");
#endif
}

// Low 32 bits of a generic (flat) shared pointer are the LDS byte address
// (ISA §10.2, LDS aperture).
__device__ __forceinline__ unsigned int lds_byte_addr(const void* p)
{
    return (unsigned int)(unsigned long long)(uintptr_t)p;
}

// Block: blockDim.x = BX channel-threads, blockDim.y = COLS columns.
// Phase 0: async-stage the block's input slab (img_x/y/z, contiguous along k) into LDS.
// Phase 1: geometry once per voxel -> (vz, intensity) pairs in LDS.
// Phase 2: each thread sweeps its column accumulating velocity channels (v_exp_f32).
__global__ __launch_bounds__(256) void cube_sim_kernel(
    const float* __restrict__ imgx, const float* __restrict__ imgy,
    const float* __restrict__ imgz,
    const float* __restrict__ p_incl, const float* __restrict__ p_skyrot,
    const float* __restrict__ p_vmin, const float* __restrict__ p_vmax,
    const float* __restrict__ p_lb,   const float* __restrict__ p_vrot,
    const float* __restrict__ p_rt,   const float* __restrict__ p_I0,
    const float* __restrict__ p_rd,   const float* __restrict__ p_hz,
    float* __restrict__ out, int G, int V)
{
    extern __shared__ __align__(16) float smem_raw[];

    const int COLS     = blockDim.y;
    const int nthreads = blockDim.x * blockDim.y;
    const int tid      = threadIdx.y * blockDim.x + threadIdx.x;
    const int GG       = G * G;
    const int colBase  = blockIdx.x * COLS;
    const int nvox     = COLS * G;

    float2* res   = reinterpret_cast<float2*>(smem_raw); // [nvox] of (vz, I)
    float*  stage = smem_raw + 2 * nvox;                 // [3 * nvox] staged x|y|z

    // Async staging legal only if slab is 16B-granular and fully in-bounds.
    const bool useAsync = ((nvox & 3) == 0) && (colBase + COLS <= GG);

    // CDNA5 prefetch of this block's contiguous input slab (global_prefetch_b8).
    {
        const size_t base = (size_t)colBase * (size_t)G;
        __builtin_prefetch(imgx + base, 0, 0);
        __builtin_prefetch(imgy + base, 0, 0);
        __builtin_prefetch(imgz + base, 0, 0);
    }

    if (useAsync) {
        const size_t base   = (size_t)colBase * (size_t)G;      // element offset
        const int    chunks = nvox >> 2;                        // float4 chunks per array
        const unsigned long long gx = (unsigned long long)(uintptr_t)(imgx + base);
        const unsigned long long gy = (unsigned long long)(uintptr_t)(imgy + base);
        const unsigned long long gz = (unsigned long long)(uintptr_t)(imgz + base);
        const unsigned int sx = lds_byte_addr(stage);
        const unsigned int sy = sx + (unsigned int)(nvox * 4);
        const unsigned int sz = sy + (unsigned int)(nvox * 4);
        for (int i = tid; i < chunks; i += nthreads) {
            const unsigned int b = (unsigned int)i * 16u;
            async_g2l_b128(sx + b, gx + b);
            async_g2l_b128(sy + b, gy + b);
            async_g2l_b128(sz + b, gz + b);
        }
        wait_asynccnt0();     // this wave's async copies complete
    }
    __syncthreads();          // all waves' copies visible block-wide

    // ---- scalar parameters (uniform loads, hit scalar cache) ----
    const float incl  = p_incl[0];
    const float skyr  = p_skyrot[0];
    const float ci = cosf(incl),  si = sinf(incl);
    const float cr = cosf(skyr),  sr = sinf(skyr);
    const float rt     = p_rt[0];
    const float rt2    = rt * rt;
    const float inv_rd = 1.0f / p_rd[0];
    const float inv_hz = 1.0f / p_hz[0];
    const float I0     = p_I0[0];
    const float svm    = -si * p_vrot[0];           // v_z_r = -si*v_max*rot_x*rsqrt(r^2+rt^2)
    const float lb     = p_lb[0];
    const float sigsq  = lb * lb;
    const float nc     = -(1.0f / sigsq) * LOG2E_F; // exp(-d^2/s2) = exp2(d^2 * nc)
    const float vmin_pc = p_vmin[0] / PC_F;
    const float vmax_pc = p_vmax[0] / PC_F;
    const float lstep   = (V > 1) ? (vmax_pc - vmin_pc) / (float)(V - 1) : 0.0f;
    const float norm    = 1.0f / sqrtf(6.2831853071795864f * sigsq);

    // ---- phase 1: geometry, once per voxel ----
    for (int v = tid; v < nvox; v += nthreads) {
        float x, y, z;
        if (useAsync) {
            x = stage[v];
            y = stage[nvox + v];
            z = stage[2 * nvox + v];
        } else {
            const int c   = v / G;
            const int k   = v - c * G;
            const int col = colBase + c;
            x = y = z = 0.0f;
            if (col < GG) {
                const size_t idx = (size_t)col * (size_t)G + (size_t)k;
                x = imgx[idx]; y = imgy[idx]; z = imgz[idx];
            }
        }
        const float x1 = x * cr - y * sr;           // rot_x (Rx leaves x unchanged)
        const float y1 = x * sr + y * cr;
        const float ry = y1 * ci - z * si;          // rot_y
        const float rz = y1 * si + z * ci;          // rot_z
        const float r2 = x1 * x1 + ry * ry;
        const float r  = sqrtf(r2);
        const float den2 = r2 + rt2;
        // v_abs*cos(atan2(ry,x1)) == v_max*x1*rsqrt(r2+rt2); guard matches nan_to_num->0
        const float vz = (den2 > 0.0f) ? (svm * x1 * rsqrtf(den2)) : 0.0f;
        const float inten = I0 * expf(-(r * inv_rd + fabsf(rz) * inv_hz));
        res[v] = make_float2(vz, inten);
    }
    __syncthreads();

    // ---- phase 2: channel sweep over staged column (ds_load_b128 broadcast) ----
    const int c   = threadIdx.y;
    const int col = colBase + c;
    if (col >= GG) return;
    const float2* colp = res + c * G;

    for (int vch = threadIdx.x; vch < V; vch += blockDim.x) {
        const float lab = vmin_pc + lstep * (float)vch;
        float acc = 0.0f;
        if ((G & 1) == 0) {
            const float4* colp4 = reinterpret_cast<const float4*>(colp);
            const int half = G >> 1;
#pragma unroll 4
            for (int q = 0; q < half; ++q) {
                const float4 p = colp4[q];            // two (vz, I) pairs per LDS b128
                const float dA = lab - p.x;
                const float dB = lab - p.z;
                const float eA = exp2f(dA * dA * nc); // single v_exp_f32 each
                const float eB = exp2f(dB * dB * nc);
                acc = fmaf(eA, p.y, acc);
                acc = fmaf(eB, p.w, acc);
            }
        } else {
            for (int k = 0; k < G; ++k) {
                const float2 p = colp[k];
                const float d = lab - p.x;
                acc = fmaf(exp2f(d * d * nc), p.y, acc);
            }
        }
        out[(size_t)vch * (size_t)GG + (size_t)col] = acc * norm;
    }
}

extern "C" void kernel_launch(void* const* d_in, const int* in_sizes, int n_in,
                              void* d_out, int out_size, void* d_ws, size_t ws_size,
                              hipStream_t stream) {
    (void)d_ws; (void)ws_size; (void)n_in;

    // Recover G from |img_x| = G^3 and V from out_size = V*G*G (host-side, no sync).
    const long n = (long)in_sizes[0];
    int G = (int)lround(cbrt((double)n));
    while ((long)G * G * G < n) ++G;
    while (G > 1 && (long)G * G * G > n) --G;
    const int GG = G * G;
    int V = (GG > 0) ? out_size / GG : 1;
    if (V < 1) V = 1;

    // Block shape: BX channel-threads x COLS columns, <=256 threads.
    int BX = (V < 64) ? V : 64;
    int COLS = 256 / BX;
    if (COLS < 1) COLS = 1;
    if (COLS > GG) COLS = GG;
    // LDS: result pairs (8B/voxel) + async staging (12B/voxel) = 20B/voxel.
    while (COLS > 1 && (size_t)COLS * (size_t)G * 20u > 65536) --COLS;
    const size_t lds_bytes = (size_t)COLS * (size_t)G * 20u;

    const int grid = (GG + COLS - 1) / COLS;
    dim3 block((unsigned)BX, (unsigned)COLS, 1);

    cube_sim_kernel<<<grid, block, lds_bytes, stream>>>(
        (const float*)d_in[0], (const float*)d_in[1], (const float*)d_in[2],
        (const float*)d_in[3], (const float*)d_in[4], (const float*)d_in[5],
        (const float*)d_in[6], (const float*)d_in[7], (const float*)d_in[8],
        (const float*)d_in[9], (const float*)d_in[10], (const float*)d_in[11],
        (const float*)d_in[12],
        (float*)d_out, G, V);
}